// DGMNetT_30537217475179
// MI455X (gfx1250) — compile-verified
//
#include <hip/hip_runtime.h>

// DGM network forward pass, fully fused, fp32 WMMA (V_WMMA_F32_16X16X4_F32).
// N = 131072 rows, H = 256 hidden, 3 layers.
// One 256-thread block (8 wave32) processes a 16-row slab; S/S1/S*R tiles in LDS.
// tanh uses the gfx1250 hardware V_TANH_F32 transcendental when available.

typedef float v2f __attribute__((ext_vector_type(2)));
typedef float v8f __attribute__((ext_vector_type(8)));

#define H       256
#define LAYERS  3
#define MBLK    16
#define STRIDE  264   // LDS row stride in floats (multiple of 8, offsets banks by 8)

__device__ __forceinline__ v8f wmma_f32x4(v2f a, v2f b, v8f c) {
    // D(16x16,f32) = A(16x4,f32) * B(4x16,f32) + C
    return __builtin_amdgcn_wmma_f32_16x16x4_f32(
        /*neg_a=*/false, a, /*neg_b=*/false, b,
        /*c_mod=*/(short)0, c, /*reuse_a=*/false, /*reuse_b=*/false);
}

__device__ __forceinline__ float fast_tanh(float v) {
#if __has_builtin(__builtin_amdgcn_tanhf)
    // gfx1250 hardware transcendental: single V_TANH_F32, branch-free.
    return __builtin_amdgcn_tanhf(v);
#else
    // Branch-free fallback: tanh(x) = 1 - 2/(exp(2x)+1).
    // x -> +inf: exp -> inf -> 1;  x -> -inf: exp -> 0 -> -1.  No divergence.
    float e = __expf(2.0f * v);
    return 1.0f - 2.0f / (e + 1.0f);
#endif
}

__global__ __launch_bounds__(256) void dgm_fused_kernel(
    const float* __restrict__ t,  const float* __restrict__ x,
    const float* __restrict__ W0, const float* __restrict__ b0,
    const float* __restrict__ Uz, const float* __restrict__ Ug,
    const float* __restrict__ Ur, const float* __restrict__ Uh,
    const float* __restrict__ Wz, const float* __restrict__ Wg,
    const float* __restrict__ Wr, const float* __restrict__ Wh,
    const float* __restrict__ bz, const float* __restrict__ bg,
    const float* __restrict__ br, const float* __restrict__ bh,
    const float* __restrict__ Wf, const float* __restrict__ bf,
    float* __restrict__ out)
{
    __shared__ float sS [MBLK][STRIDE];   // current state S
    __shared__ float sS1[MBLK][STRIDE];   // initial state S1 (fixed across layers)
    __shared__ float sSR[MBLK][STRIDE];   // S * R scratch
    __shared__ float sXt[MBLK];
    __shared__ float sXx[MBLK];

    const int tid  = threadIdx.x;
    const int wave = tid >> 5;          // 0..7
    const int lane = tid & 31;
    const int half = lane >> 4;         // 0: lanes 0-15, 1: lanes 16-31
    const int l16  = lane & 15;
    const int row0 = blockIdx.x * MBLK;

    // ---- load X tile (t, x) ----
    if (tid < MBLK) {
        sXt[tid] = t[row0 + tid];
        sXx[tid] = x[row0 + tid];
    }
    __syncthreads();

    // ---- initial dense: S1 = tanh(X @ W0 + b0); S = S1 ----
    {
        const int   c   = tid;          // 256 threads == 256 columns
        const float w0t = W0[c];
        const float w0x = W0[H + c];
        const float bb  = b0[c];
        #pragma unroll
        for (int r = 0; r < MBLK; ++r) {
            float v = fast_tanh(sXt[r] * w0t + sXx[r] * w0x + bb);
            sS1[r][c] = v;
            sS [r][c] = v;
        }
    }
    __syncthreads();

    // Per-lane X rows for accumulator init. C/D layout: element (v,lane) is
    // m = half*8 + v, n = colbase + (lane&15).
    float xt8[8], xx8[8];
    #pragma unroll
    for (int v = 0; v < 8; ++v) {
        xt8[v] = sXt[half * 8 + v];
        xx8[v] = sXx[half * 8 + v];
    }

    const int koff = half * 2;   // A/B fragment K sub-offset (lanes 16-31 hold K+2,K+3)
    const int cb0  = wave * 32;  // two 16-wide column tiles per wave

    for (int l = 0; l < LAYERS; ++l) {
        const float* wzl = Wz + l * H * H;
        const float* wgl = Wg + l * H * H;
        const float* wrl = Wr + l * H * H;
        const float* whl = Wh + l * H * H;
        const float* uzl = Uz + l * 2 * H;
        const float* ugl = Ug + l * 2 * H;
        const float* url = Ur + l * 2 * H;
        const float* uhl = Uh + l * 2 * H;
        const float* bzl = bz + l * H;
        const float* bgl = bg + l * H;
        const float* brl = br + l * H;
        const float* bhl = bh + l * H;

        // ================= phase 1: R gate, SR = S * R =================
        v8f accR[2];
        #pragma unroll
        for (int t2 = 0; t2 < 2; ++t2) {
            const int   n  = cb0 + t2 * 16 + l16;
            const float u0 = url[n];
            const float u1 = url[H + n];
            const float bb = brl[n];
            #pragma unroll
            for (int v = 0; v < 8; ++v)
                accR[t2][v] = xt8[v] * u0 + xx8[v] * u1 + bb;   // X@Ur + br
        }
        for (int k = 0; k < H; k += 4) {
            // A fragment (16x4 f32): lane row = l16, K pair at k+koff
            v2f a = *(const v2f*)&sS[l16][k + koff];
            #pragma unroll
            for (int t2 = 0; t2 < 2; ++t2) {
                const int n = cb0 + t2 * 16 + l16;
                v2f b;
                b.x = wrl[(k + koff)     * H + n];
                b.y = wrl[(k + koff + 1) * H + n];
                accR[t2] = wmma_f32x4(a, b, accR[t2]);
            }
        }
        #pragma unroll
        for (int t2 = 0; t2 < 2; ++t2) {
            const int n = cb0 + t2 * 16 + l16;
            #pragma unroll
            for (int v = 0; v < 8; ++v) {
                const int m = half * 8 + v;
                float R = fast_tanh(accR[t2][v]);
                sSR[m][n] = sS[m][n] * R;
            }
        }
        __syncthreads();

        // ============ phase 2: Z = tanh(X@Uz + S@Wz + bz)
        //                        G = tanh(X@Ug + S1@Wg + bg)
        //                        Hg = tanh(X@Uh + SR@Wh + bh) ============
        v8f accZ[2], accG[2], accHg[2];
        #pragma unroll
        for (int t2 = 0; t2 < 2; ++t2) {
            const int   n  = cb0 + t2 * 16 + l16;
            const float z0 = uzl[n], z1 = uzl[H + n], zb = bzl[n];
            const float g0 = ugl[n], g1 = ugl[H + n], gb = bgl[n];
            const float h0 = uhl[n], h1 = uhl[H + n], hb = bhl[n];
            #pragma unroll
            for (int v = 0; v < 8; ++v) {
                accZ [t2][v] = xt8[v] * z0 + xx8[v] * z1 + zb;
                accG [t2][v] = xt8[v] * g0 + xx8[v] * g1 + gb;
                accHg[t2][v] = xt8[v] * h0 + xx8[v] * h1 + hb;
            }
        }
        for (int k = 0; k < H; k += 4) {
            v2f aS  = *(const v2f*)&sS [l16][k + koff];
            v2f aS1 = *(const v2f*)&sS1[l16][k + koff];
            v2f aSR = *(const v2f*)&sSR[l16][k + koff];
            #pragma unroll
            for (int t2 = 0; t2 < 2; ++t2) {
                const int n = cb0 + t2 * 16 + l16;
                v2f bzv, bgv, bhv;
                bzv.x = wzl[(k + koff)     * H + n];
                bzv.y = wzl[(k + koff + 1) * H + n];
                bgv.x = wgl[(k + koff)     * H + n];
                bgv.y = wgl[(k + koff + 1) * H + n];
                bhv.x = whl[(k + koff)     * H + n];
                bhv.y = whl[(k + koff + 1) * H + n];
                accZ [t2] = wmma_f32x4(aS,  bzv, accZ [t2]);
                accG [t2] = wmma_f32x4(aS1, bgv, accG [t2]);
                accHg[t2] = wmma_f32x4(aSR, bhv, accHg[t2]);
            }
        }
        // Snew = (1 - G) * Hg + Z * S  (into registers first; S is still being
        // read by other waves' K-loops until the barrier)
        float snew[2][8];
        #pragma unroll
        for (int t2 = 0; t2 < 2; ++t2) {
            const int n = cb0 + t2 * 16 + l16;
            #pragma unroll
            for (int v = 0; v < 8; ++v) {
                const int m = half * 8 + v;
                float Zv = fast_tanh(accZ [t2][v]);
                float Gv = fast_tanh(accG [t2][v]);
                float Hv = fast_tanh(accHg[t2][v]);
                snew[t2][v] = (1.0f - Gv) * Hv + Zv * sS[m][n];
            }
        }
        __syncthreads();
        #pragma unroll
        for (int t2 = 0; t2 < 2; ++t2) {
            const int n = cb0 + t2 * 16 + l16;
            #pragma unroll
            for (int v = 0; v < 8; ++v)
                sS[half * 8 + v][n] = snew[t2][v];
        }
        __syncthreads();
    }

    // ---- final projection: out = S @ Wf + bf  (16 lanes per row) ----
    {
        const int r  = tid >> 4;       // 0..15
        const int ci = tid & 15;       // 16-way column split
        float p = 0.0f;
        #pragma unroll
        for (int j = 0; j < 16; ++j) {
            const int c = ci * 16 + j;
            p += sS[r][c] * Wf[c];
        }
        #pragma unroll
        for (int off = 8; off >= 1; off >>= 1)
            p += __shfl_xor(p, off, 16);
        if (ci == 0)
            out[row0 + r] = p + bf[0];
    }
}

extern "C" void kernel_launch(void* const* d_in, const int* in_sizes, int n_in,
                              void* d_out, int out_size, void* d_ws, size_t ws_size,
                              hipStream_t stream) {
    (void)n_in; (void)d_ws; (void)ws_size; (void)out_size;
    const float* t  = (const float*)d_in[0];
    const float* x  = (const float*)d_in[1];
    const float* W0 = (const float*)d_in[2];
    const float* b0 = (const float*)d_in[3];
    const float* Uz = (const float*)d_in[4];
    const float* Ug = (const float*)d_in[5];
    const float* Ur = (const float*)d_in[6];
    const float* Uh = (const float*)d_in[7];
    const float* Wz = (const float*)d_in[8];
    const float* Wg = (const float*)d_in[9];
    const float* Wr = (const float*)d_in[10];
    const float* Wh = (const float*)d_in[11];
    const float* bz = (const float*)d_in[12];
    const float* bg = (const float*)d_in[13];
    const float* br = (const float*)d_in[14];
    const float* bh = (const float*)d_in[15];
    const float* Wf = (const float*)d_in[16];
    const float* bf = (const float*)d_in[17];
    float* out = (float*)d_out;

    const int N = in_sizes[0];          // 131072 (multiple of MBLK)
    const int blocks = N / MBLK;
    dgm_fused_kernel<<<blocks, 256, 0, stream>>>(
        t, x, W0, b0, Uz, Ug, Ur, Uh, Wz, Wg, Wr, Wh,
        bz, bg, br, bh, Wf, bf, out);
}